// Encoder_58136677318977
// MI455X (gfx1250) — compile-verified
//
#include <hip/hip_runtime.h>
#include <hip/hip_bf16.h>
#include <stdint.h>

// Problem constants (match reference)
#define NG   50000
#define ND   20000
#define DIN  128
#define DH   64
#define EGG  1500000
#define EGD  1000000
#define EPSV 1e-5f
#define SLOPE 0.01f

#define KC   32   // K-chunk staged in LDS
#define KCP  36   // padded LDS stride (8B-aligned frags, conflict-free)

typedef __attribute__((ext_vector_type(2))) float v2f;
typedef __attribute__((ext_vector_type(8))) float v8f;

// ---------------------------------------------------------------------------
// Edge in-degree counts (mean denominator)
// ---------------------------------------------------------------------------
__global__ __launch_bounds__(256) void count_edges_kernel(
    const int* __restrict__ dst, float* __restrict__ cnt, int E) {
  int e = blockIdx.x * blockDim.x + threadIdx.x;
  if (e < E) atomicAdd(&cnt[dst[e]], 1.0f);
}

// ---------------------------------------------------------------------------
// Edge feature scatter-add: agg[dst] += x[src]; float4 gather, scalar atomics.
// qshift = log2(din/4); (256>>qshift) edges per block.
// ---------------------------------------------------------------------------
__global__ __launch_bounds__(256) void scatter_feats_kernel(
    const float* __restrict__ xsrc, const int* __restrict__ src,
    const int* __restrict__ dst, float* __restrict__ agg, int E, int qshift) {
  const int dinq = 1 << qshift;
  const int e = blockIdx.x * (256 >> qshift) + (threadIdx.x >> qshift);
  if (e >= E) return;
  const int q = threadIdx.x & (dinq - 1);
  const int s = src[e], d = dst[e];
  float4 v = reinterpret_cast<const float4*>(xsrc)[(long long)s * dinq + q];
  float* dp = agg + ((long long)d * dinq + q) * 4;
  atomicAdd(dp + 0, v.x);
  atomicAdd(dp + 1, v.y);
  atomicAdd(dp + 2, v.z);
  atomicAdd(dp + 3, v.w);
}

// ---------------------------------------------------------------------------
// Fused SAGE GEMM via V_WMMA_F32_16X16X4_F32, templated on term config so the
// K-loop is branch-free:
//   NMEAN=2,SUMW2=1 : out = scale*(A0/c0@W0 + A1/c1@W1 + Ap@(Wpa+Wpb) + b0+b1)
//   NMEAN=1,SUMW2=0 : out = scale*(A0/c0@W0 + Ap@Wpa + b0)
//   NMEAN=0,SUMW2=0 : out = scale*(Ap@Wpa + b0)
// Weights are staged per K-chunk in LDS *transposed* (ldsW[t][col][k]) so a
// B fragment is one aligned 8-byte LDS read; Wpa+Wpb is folded at staging.
// One wave = 16 rows x 64 cols. EXEC stays all-ones around WMMA (uniform
// control flow; A rows clamped, stores predicated).
// A frag (16x4 f32): lanes 0-15 hold K=k0,k0+1; lanes 16-31 hold K=k0+2,k0+3.
// B frag (4x16 f32): n = lane&15; same K split by lane half.
// C/D (16x16 f32):   elem i of lane L -> row m0+i+8*(L>=16), col L&15.
// ---------------------------------------------------------------------------
template <int NMEAN, bool SUMW2>
__global__ __launch_bounds__(128) void gemm_wmma_kernel(
    const float* __restrict__ A0, const float* __restrict__ cnt0, const float* __restrict__ W0,
    const float* __restrict__ A1, const float* __restrict__ cnt1, const float* __restrict__ W1,
    const float* __restrict__ Ap, const float* __restrict__ Wpa, const float* __restrict__ Wpb,
    const float* __restrict__ bias0, const float* __restrict__ bias1,
    float scale, float* __restrict__ out, int N, int din) {
  constexpr int NT = NMEAN + 1;
  __shared__ float ldsW[NT][DH][KCP];

  const int lane = threadIdx.x & 31;
  const int wave = threadIdx.x >> 5;
  const int m0   = (blockIdx.x * 4 + wave) * 16;
  const int half = lane >> 4;   // 0: K{0,1}, 1: K{2,3}
  const int l16  = lane & 15;

  v8f z = {};
  v8f acc[4];
  acc[0] = z; acc[1] = z; acc[2] = z; acc[3] = z;

  int rowA = m0 + l16;
  if (rowA >= N) rowA = N - 1;            // clamp; garbage rows masked at store
  const long long rb = (long long)rowA * din;

  float rcp0 = 1.0f, rcp1 = 1.0f;
  if constexpr (NMEAN >= 1) rcp0 = 1.0f / fmaxf(cnt0[rowA], 1.0f);
  if constexpr (NMEAN >= 2) rcp1 = 1.0f / fmaxf(cnt1[rowA], 1.0f);

  for (int kc = 0; kc < din; kc += KC) {
    __syncthreads();                      // previous chunk fully consumed
    // ---- stage weights (transposed, Wpa+Wpb folded), coalesced global reads
    for (int e = threadIdx.x; e < KC * DH; e += 128) {
      const int kk  = e >> 6;             // 0..KC-1
      const int col = e & (DH - 1);
      const int gk  = kc + kk;
      if constexpr (NMEAN >= 1) ldsW[0][col][kk] = W0[gk * DH + col];
      if constexpr (NMEAN >= 2) ldsW[1][col][kk] = W1[gk * DH + col];
      float wp = Wpa[gk * DH + col];
      if constexpr (SUMW2) wp += Wpb[gk * DH + col];
      ldsW[NMEAN][col][kk] = wp;
    }
    __syncthreads();

    // ---- prefetch next A chunk (global_prefetch_b8) while we crunch this one
    if (kc + KC < din) {
      if constexpr (NMEAN >= 1) __builtin_prefetch(A0 + rb + kc + KC, 0, 0);
      if constexpr (NMEAN >= 2) __builtin_prefetch(A1 + rb + kc + KC, 0, 0);
      __builtin_prefetch(Ap + rb + kc + KC, 0, 0);
    }

#pragma unroll
    for (int k0 = 0; k0 < KC; k0 += 4) {
      const int kl = k0 + half * 2;       // local K for this lane's fragment
      const long long gi = rb + kc + kl;
      if constexpr (NMEAN >= 1) {
        v2f a; a.x = A0[gi] * rcp0; a.y = A0[gi + 1] * rcp0;
#pragma unroll
        for (int t = 0; t < 4; ++t) {
          v2f b = *(const v2f*)&ldsW[0][t * 16 + l16][kl];
          acc[t] = __builtin_amdgcn_wmma_f32_16x16x4_f32(
              false, a, false, b, (short)0, acc[t], false, false);
        }
      }
      if constexpr (NMEAN >= 2) {
        v2f a; a.x = A1[gi] * rcp1; a.y = A1[gi + 1] * rcp1;
#pragma unroll
        for (int t = 0; t < 4; ++t) {
          v2f b = *(const v2f*)&ldsW[1][t * 16 + l16][kl];
          acc[t] = __builtin_amdgcn_wmma_f32_16x16x4_f32(
              false, a, false, b, (short)0, acc[t], false, false);
        }
      }
      {
        v2f a; a.x = Ap[gi]; a.y = Ap[gi + 1];
#pragma unroll
        for (int t = 0; t < 4; ++t) {
          v2f b = *(const v2f*)&ldsW[NMEAN][t * 16 + l16][kl];
          acc[t] = __builtin_amdgcn_wmma_f32_16x16x4_f32(
              false, a, false, b, (short)0, acc[t], false, false);
        }
      }
    }
  }

#pragma unroll
  for (int t = 0; t < 4; ++t) {
    const int col = t * 16 + l16;
    float bt = bias0[col];
    if (bias1) bt += bias1[col];
#pragma unroll
    for (int i = 0; i < 8; ++i) {
      int row = m0 + i + half * 8;
      if (row < N) out[(long long)row * DH + col] = (acc[t][i] + bt) * scale;
    }
  }
}

// ---------------------------------------------------------------------------
// BatchNorm pass 1: per-column sum / sumsq (64 cols) via block partials.
// stats[0:64] = sum, stats[64:128] = sumsq. Must be zeroed beforehand.
// ---------------------------------------------------------------------------
__global__ __launch_bounds__(256) void bn_stats_kernel(
    const float* __restrict__ x, float* __restrict__ stats, int N) {
  __shared__ float ls[256], lq[256];
  const int col = threadIdx.x & 63;
  const int rg  = threadIdx.x >> 6;          // 0..3 row groups
  float s = 0.f, q = 0.f;
  for (int r = blockIdx.x * 4 + rg; r < N; r += gridDim.x * 4) {
    float v = x[(long long)r * DH + col];
    s += v; q += v * v;
  }
  ls[threadIdx.x] = s; lq[threadIdx.x] = q;
  __syncthreads();
  if (threadIdx.x < 64) {
    float S = ls[threadIdx.x] + ls[threadIdx.x + 64] + ls[threadIdx.x + 128] + ls[threadIdx.x + 192];
    float Q = lq[threadIdx.x] + lq[threadIdx.x + 64] + lq[threadIdx.x + 128] + lq[threadIdx.x + 192];
    atomicAdd(&stats[threadIdx.x], S);
    atomicAdd(&stats[DH + threadIdx.x], Q);
  }
}

// ---------------------------------------------------------------------------
// BatchNorm pass 2 + LeakyReLU + optional residual.
// ---------------------------------------------------------------------------
__global__ __launch_bounds__(256) void bn_apply_kernel(
    const float* __restrict__ x, const float* __restrict__ stats,
    const float* __restrict__ g, const float* __restrict__ b,
    const float* __restrict__ res, float* __restrict__ out, int N, float invN) {
  long long i = (long long)blockIdx.x * blockDim.x + threadIdx.x;
  if (i >= (long long)N * DH) return;
  int col = (int)(i & (DH - 1));
  float m   = stats[col] * invN;
  float var = stats[DH + col] * invN - m * m;
  float y = (x[i] - m) * rsqrtf(var + EPSV) * g[col] + b[col];
  y = (y >= 0.f) ? y : SLOPE * y;
  if (res) y += res[i];
  out[i] = y;
}

// ---------------------------------------------------------------------------
// Host orchestration
// ---------------------------------------------------------------------------
static inline unsigned cdiv(long long a, long long b) { return (unsigned)((a + b - 1) / b); }

extern "C" void kernel_launch(void* const* d_in, const int* in_sizes, int n_in,
                              void* d_out, int out_size, void* d_ws, size_t ws_size,
                              hipStream_t stream) {
  (void)in_sizes; (void)n_in; (void)out_size; (void)ws_size;

  const float* x_gene = (const float*)d_in[0];  // [NG,128]
  const float* x_dis  = (const float*)d_in[1];  // [ND,128]
  const int* egg = (const int*)d_in[2];         // [2,EGG]
  const int* egd = (const int*)d_in[3];         // [2,EGD]
  const int* edg = (const int*)d_in[4];         // [2,EGD]
  const int *egg_src = egg,       *egg_dst = egg + EGG;
  const int *egd_src = egd,       *egd_dst = egd + EGD;
  const int *edg_src = edg,       *edg_dst = edg + EGD;

  auto F = [&](int i) { return (const float*)d_in[i]; };

  // params flat-leaf indices (pytree alphabetical order inside dicts)
  // layer li base L = 5 + 13*li:
  //  +0 bn_dis.b +1 bn_dis.g +2 bn_gene.b +3 bn_gene.g
  //  +4 dg.Wl +5 dg.Wr +6 dg.b  +7 gd.Wl +8 gd.Wr +9 gd.b
  //  +10 gg.Wl +11 gg.Wr +12 gg.b
  // post base 44: dis{bn.b,bn.g,l1.W,l1.b,l2.W,l2.b} gene{same}

  // ---- workspace carve (floats) ----
  float* ws = (float*)d_ws;
  size_t off = 0;
  auto alloc = [&](size_t n) { float* p = ws + off; off += (n + 63) & ~(size_t)63; return p; };
  float* agg_g   = alloc((size_t)NG * DIN);   // gg aggregation (max din)
  float* agg_g2  = alloc((size_t)NG * DIN);   // dg aggregation
  float* agg_d   = alloc((size_t)ND * DIN);   // gd aggregation
  float* cnt_g   = alloc(NG);
  float* cnt_g2  = alloc(NG);
  float* cnt_d   = alloc(ND);
  float* hg_raw  = alloc((size_t)NG * DH);
  float* hg_pp0  = alloc((size_t)NG * DH);
  float* hg_pp1  = alloc((size_t)NG * DH);
  float* hd_raw  = alloc((size_t)ND * DH);
  float* hd_pp0  = alloc((size_t)ND * DH);
  float* hd_pp1  = alloc((size_t)ND * DH);
  float* stats_g = alloc(2 * DH);
  float* stats_d = alloc(2 * DH);

  float* out_g = (float*)d_out;                       // [NG,64]
  float* out_d = (float*)d_out + (size_t)NG * DH;     // [ND,64]

  // ---- in-degree counts (same edges every layer: compute once) ----
  hipMemsetAsync(cnt_g,  0, NG * sizeof(float), stream);
  hipMemsetAsync(cnt_g2, 0, NG * sizeof(float), stream);
  hipMemsetAsync(cnt_d,  0, ND * sizeof(float), stream);
  count_edges_kernel<<<cdiv(EGG, 256), 256, 0, stream>>>(egg_dst, cnt_g, EGG);
  count_edges_kernel<<<cdiv(EGD, 256), 256, 0, stream>>>(edg_dst, cnt_g2, EGD);
  count_edges_kernel<<<cdiv(EGD, 256), 256, 0, stream>>>(egd_dst, cnt_d, EGD);

  auto bn_block = [&](const float* raw, float* stats, const float* g, const float* b,
                      const float* res, float* outp, int N) {
    hipMemsetAsync(stats, 0, 2 * DH * sizeof(float), stream);
    bn_stats_kernel<<<256, 256, 0, stream>>>(raw, stats, N);
    bn_apply_kernel<<<cdiv((long long)N * DH, 256), 256, 0, stream>>>(
        raw, stats, g, b, res, outp, N, 1.0f / (float)N);
  };

  const float* xg = x_gene;
  const float* xd = x_dis;
  float* g_out_buf[3] = {hg_pp0, hg_pp1, hg_pp0};
  float* d_out_buf[3] = {hd_pp0, hd_pp1, hd_pp0};

  for (int li = 0; li < 3; ++li) {
    const int L   = 5 + 13 * li;
    const int din = (li == 0) ? DIN : DH;
    const int qshift = (din == DIN) ? 5 : 4;    // log2(din/4)
    const int epb = 256 >> qshift;              // edges per block
    const float *dg_Wl = F(L+4),  *dg_Wr = F(L+5),  *dg_b = F(L+6);
    const float *gd_Wl = F(L+7),  *gd_Wr = F(L+8),  *gd_b = F(L+9);
    const float *gg_Wl = F(L+10), *gg_Wr = F(L+11), *gg_b = F(L+12);

    hipMemsetAsync(agg_g,  0, (size_t)NG * din * sizeof(float), stream);
    hipMemsetAsync(agg_g2, 0, (size_t)NG * din * sizeof(float), stream);
    hipMemsetAsync(agg_d,  0, (size_t)ND * din * sizeof(float), stream);
    scatter_feats_kernel<<<cdiv(EGG, epb), 256, 0, stream>>>(
        xg, egg_src, egg_dst, agg_g, EGG, qshift);
    scatter_feats_kernel<<<cdiv(EGD, epb), 256, 0, stream>>>(
        xd, edg_src, edg_dst, agg_g2, EGD, qshift);
    scatter_feats_kernel<<<cdiv(EGD, epb), 256, 0, stream>>>(
        xg, egd_src, egd_dst, agg_d, EGD, qshift);

    // gene: 0.5*( mean_gg@Wl_gg + mean_dg@Wl_dg + xg@(Wr_gg+Wr_dg) + b_gg+b_dg )
    gemm_wmma_kernel<2, true><<<cdiv(NG, 64), 128, 0, stream>>>(
        agg_g, cnt_g, gg_Wl, agg_g2, cnt_g2, dg_Wl, xg, gg_Wr, dg_Wr,
        gg_b, dg_b, 0.5f, hg_raw, NG, din);
    // disease: mean_gd@Wl_gd + xd@Wr_gd + b_gd
    gemm_wmma_kernel<1, false><<<cdiv(ND, 64), 128, 0, stream>>>(
        agg_d, cnt_d, gd_Wl, nullptr, nullptr, nullptr, xd, gd_Wr, nullptr,
        gd_b, nullptr, 1.0f, hd_raw, ND, din);

    const float* res_g = (li == 0) ? nullptr : xg;
    const float* res_d = (li == 0) ? nullptr : xd;
    bn_block(hg_raw, stats_g, F(L+3), F(L+2), res_g, g_out_buf[li], NG);  // bn_gene {g=+3,b=+2}
    bn_block(hd_raw, stats_d, F(L+1), F(L+0), res_d, d_out_buf[li], ND);  // bn_dis  {g=+1,b=+0}
    xg = g_out_buf[li];
    xd = d_out_buf[li];
  }

  // ---- post MLPs: lin1 -> BN -> lrelu -> lin2 ----
  // dis: 44 bn.b, 45 bn.g, 46 lin1.W, 47 lin1.b, 48 lin2.W, 49 lin2.b
  // gene:50 bn.b, 51 bn.g, 52 lin1.W, 53 lin1.b, 54 lin2.W, 55 lin2.b
  // gene head
  gemm_wmma_kernel<0, false><<<cdiv(NG, 64), 128, 0, stream>>>(
      nullptr, nullptr, nullptr, nullptr, nullptr, nullptr, xg, F(52), nullptr,
      F(53), nullptr, 1.0f, hg_raw, NG, DH);
  bn_block(hg_raw, stats_g, F(51), F(50), nullptr, hg_pp1, NG);
  gemm_wmma_kernel<0, false><<<cdiv(NG, 64), 128, 0, stream>>>(
      nullptr, nullptr, nullptr, nullptr, nullptr, nullptr, hg_pp1, F(54), nullptr,
      F(55), nullptr, 1.0f, out_g, NG, DH);
  // disease head
  gemm_wmma_kernel<0, false><<<cdiv(ND, 64), 128, 0, stream>>>(
      nullptr, nullptr, nullptr, nullptr, nullptr, nullptr, xd, F(46), nullptr,
      F(47), nullptr, 1.0f, hd_raw, ND, DH);
  bn_block(hd_raw, stats_d, F(45), F(44), nullptr, hd_pp1, ND);
  gemm_wmma_kernel<0, false><<<cdiv(ND, 64), 128, 0, stream>>>(
      nullptr, nullptr, nullptr, nullptr, nullptr, nullptr, hd_pp1, F(48), nullptr,
      F(49), nullptr, 1.0f, out_d, ND, DH);
}